// Attention_o_policy_2010044694920
// MI455X (gfx1250) — compile-verified
//
#include <hip/hip_runtime.h>
#include <hip/hip_bf16.h>

typedef float v2f __attribute__((ext_vector_type(2)));
typedef float v4f __attribute__((ext_vector_type(4)));
typedef float v8f __attribute__((ext_vector_type(8)));

// D = A(16x4 f32) * B(4x16 f32) + C(16x16 f32), full-precision CDNA5 WMMA.
#define WMMA_F32(a, b, c) \
  __builtin_amdgcn_wmma_f32_16x16x4_f32(false, (a), false, (b), (short)0, (c), false, false)

#define NUM_HEADS 12
#define HD        64
#define NTOK      1024
#define CDIM      768
#define SFEAT     768
#define EPSV      1e-6f

__device__ __forceinline__ float hmax16(float v) {
  v = fmaxf(v, __shfl_xor(v, 1, 32));
  v = fmaxf(v, __shfl_xor(v, 2, 32));
  v = fmaxf(v, __shfl_xor(v, 4, 32));
  v = fmaxf(v, __shfl_xor(v, 8, 32));
  return v;
}
__device__ __forceinline__ float hsum16(float v) {
  v += __shfl_xor(v, 1, 32);
  v += __shfl_xor(v, 2, 32);
  v += __shfl_xor(v, 4, 32);
  v += __shfl_xor(v, 8, 32);
  return v;
}

// -------------------------------------------------------------------------
// C[M][N] = X[M][K] @ W[N][K]^T + bias[N]
// 256 threads = 8 waves; block tile 128(M) x 64(N); K staged in 32-chunks.
// Each wave owns 16 M-rows and 4 16x16 N-subtiles (v8f accumulators).
// -------------------------------------------------------------------------
__global__ __launch_bounds__(256)
void gemm_xwt_bias(const float* __restrict__ X, const float* __restrict__ W,
                   const float* __restrict__ bias, float* __restrict__ C,
                   int M, int N, int K) {
  __shared__ float Xl[128 * 36];
  __shared__ float Wl[64 * 36];

  const int tid  = threadIdx.x;
  const int wave = tid >> 5;
  const int lane = tid & 31;
  const int l16  = lane & 15;
  const int h2   = lane >> 4;
  const int nb0  = blockIdx.x * 64;
  const int mb0  = blockIdx.y * 128;

  v8f acc[4];
#pragma unroll
  for (int j = 0; j < 4; ++j) {
    float bv = bias[nb0 + 16 * j + l16];
#pragma unroll
    for (int r = 0; r < 8; ++r) acc[j][r] = bv;
  }

  for (int k0 = 0; k0 < K; k0 += 32) {
    // X tile: 128 rows x 8 float4
#pragma unroll
    for (int i = 0; i < 4; ++i) {
      int f = tid + 256 * i;            // 0..1023
      int row = f >> 3, c4 = f & 7;
      v4f v = *(const v4f*)(X + (size_t)(mb0 + row) * K + k0 + c4 * 4);
      *(v4f*)&Xl[row * 36 + c4 * 4] = v;
    }
    // W tile: 64 rows x 8 float4
#pragma unroll
    for (int i = 0; i < 2; ++i) {
      int f = tid + 256 * i;            // 0..511
      int row = f >> 3, c4 = f & 7;
      v4f v = *(const v4f*)(W + (size_t)(nb0 + row) * K + k0 + c4 * 4);
      *(v4f*)&Wl[row * 36 + c4 * 4] = v;
    }
    __syncthreads();

#pragma unroll
    for (int kk = 0; kk < 32; kk += 4) {
      v2f a = *(const v2f*)&Xl[(16 * wave + l16) * 36 + kk + 2 * h2];
#pragma unroll
      for (int j = 0; j < 4; ++j) {
        v2f b = *(const v2f*)&Wl[(16 * j + l16) * 36 + kk + 2 * h2];
        acc[j] = WMMA_F32(a, b, acc[j]);
      }
    }
    __syncthreads();
  }

#pragma unroll
  for (int j = 0; j < 4; ++j) {
#pragma unroll
    for (int r = 0; r < 8; ++r) {
      int mrow = mb0 + 16 * wave + r + 8 * h2;
      int ncol = nb0 + 16 * j + l16;
      C[(size_t)mrow * N + ncol] = acc[j][r];
    }
  }
}

// -------------------------------------------------------------------------
// colsum[b][h][d] = sum_n V[b][h][n][d]  (V lives inside the qkv buffer)
// -------------------------------------------------------------------------
__global__ __launch_bounds__(64)
void v_colsum(const float* __restrict__ qkv, float* __restrict__ cs) {
  const int bh = blockIdx.x;            // 0..47
  const int b = bh / NUM_HEADS, h = bh % NUM_HEADS;
  const int d = threadIdx.x;            // 0..63
  const size_t base = (size_t)b * NTOK * 3 * CDIM + 2 * CDIM + h * HD + d;
  float s = 0.f;
  for (int n = 0; n < NTOK; ++n) s += qkv[base + (size_t)n * 3 * CDIM];
  cs[(size_t)bh * HD + d] = s;
}

// -------------------------------------------------------------------------
// Flash-style attention with one-hot policy mask + exact eps handling.
// 128 threads = 4 waves x 16 queries; grid (N/64, H, B).
// mask(n,m)=1 iff cls[n]==cls[m] || cls[n]==2 || cls[m]==2 (derived from ap).
// out = (sum e*V + (eps/N)*colsumV) / (sum e + eps), e = exp(S - rowmax)*mask
// -------------------------------------------------------------------------
__global__ __launch_bounds__(128)
void attn_policy(const float* __restrict__ qkv, const float* __restrict__ decision,
                 const float* __restrict__ colsum, float* __restrict__ out) {
  __shared__ float Ql[64 * 68];     // Q (pre-scaled)
  __shared__ float Kl[64 * 68];     // K tile; reused as P after a barrier
  __shared__ float Vt[64 * 68];     // V tile, transposed [d][key]
  __shared__ int   cls[NTOK];

  float* Pl = Kl;

  const int tid  = threadIdx.x;
  const int wave = tid >> 5;
  const int lane = tid & 31;
  const int l16  = lane & 15;
  const int h2   = lane >> 4;
  const int qt = blockIdx.x, h = blockIdx.y, b = blockIdx.z;
  const size_t qbase = (size_t)b * NTOK * 3 * CDIM;
  const float scale = 0.125f;       // 64^-0.5

  // token classes for this batch
  for (int n = tid; n < NTOK; n += 128) {
    int c = 0;
    if (n >= NTOK - SFEAT) {
      float d1 = decision[((size_t)b * SFEAT + (n - (NTOK - SFEAT))) * 2 + 1];
      c = (d1 > 0.5f) ? 2 : 1;
    }
    cls[n] = c;
  }
  // Q tile (scale folded in)
  for (int f = tid; f < 64 * 16; f += 128) {
    int row = f >> 4, c4 = f & 15;
    v4f v = *(const v4f*)(qkv + qbase + (size_t)(qt * 64 + row) * (3 * CDIM) + h * HD + c4 * 4);
    v *= scale;
    *(v4f*)&Ql[row * 68 + c4 * 4] = v;
  }
  __syncthreads();

  int clsq[8];
#pragma unroll
  for (int r = 0; r < 8; ++r) clsq[r] = cls[qt * 64 + 16 * wave + r + 8 * h2];

  v8f O[4];
  float mrow[8], lrow[8];
#pragma unroll
  for (int j = 0; j < 4; ++j)
#pragma unroll
    for (int r = 0; r < 8; ++r) O[j][r] = 0.f;
#pragma unroll
  for (int r = 0; r < 8; ++r) { mrow[r] = -INFINITY; lrow[r] = 0.f; }

  for (int kt = 0; kt < NTOK / 64; ++kt) {
    const int k0 = kt * 64;
    // load K tile (row-major) and V tile (transposed)
    for (int f = tid; f < 64 * 16; f += 128) {
      int row = f >> 4, c4 = f & 15;
      const float* src = qkv + qbase + (size_t)(k0 + row) * (3 * CDIM) + h * HD;
      v4f kv = *(const v4f*)(src + CDIM + c4 * 4);
      *(v4f*)&Kl[row * 68 + c4 * 4] = kv;
      v4f vv = *(const v4f*)(src + 2 * CDIM + c4 * 4);
      Vt[(c4 * 4 + 0) * 68 + row] = vv[0];
      Vt[(c4 * 4 + 1) * 68 + row] = vv[1];
      Vt[(c4 * 4 + 2) * 68 + row] = vv[2];
      Vt[(c4 * 4 + 3) * 68 + row] = vv[3];
    }
    __syncthreads();

    // S = (Q*scale) @ K^T  : 16 queries x 64 keys per wave
    v8f s[4];
#pragma unroll
    for (int j = 0; j < 4; ++j)
#pragma unroll
      for (int r = 0; r < 8; ++r) s[j][r] = 0.f;
#pragma unroll
    for (int kk = 0; kk < HD; kk += 4) {
      v2f a = *(const v2f*)&Ql[(16 * wave + l16) * 68 + kk + 2 * h2];
#pragma unroll
      for (int j = 0; j < 4; ++j) {
        v2f bf = *(const v2f*)&Kl[(16 * j + l16) * 68 + kk + 2 * h2];
        s[j] = WMMA_F32(a, bf, s[j]);
      }
    }
    __syncthreads();   // everyone done reading Kl before P overwrites it

    int ck[4];
#pragma unroll
    for (int j = 0; j < 4; ++j) ck[j] = cls[k0 + 16 * j + l16];

    // online softmax update (raw-S max, mask applied to e only)
#pragma unroll
    for (int r = 0; r < 8; ++r) {
      float tm = fmaxf(fmaxf(s[0][r], s[1][r]), fmaxf(s[2][r], s[3][r]));
      tm = hmax16(tm);
      float mnew = fmaxf(mrow[r], tm);
      float corr = __expf(mrow[r] - mnew);
      float rs = 0.f;
      const int cq = clsq[r];
#pragma unroll
      for (int j = 0; j < 4; ++j) {
        bool ok = (cq == ck[j]) || (cq == 2) || (ck[j] == 2);
        float e = ok ? __expf(s[j][r] - mnew) : 0.f;
        rs += e;
        Pl[(16 * wave + r + 8 * h2) * 68 + 16 * j + l16] = e;
        O[j][r] *= corr;
      }
      rs = hsum16(rs);
      lrow[r] = lrow[r] * corr + rs;
      mrow[r] = mnew;
    }

    // O += P @ V   (P as A fragments out of wave-private LDS rows)
#pragma unroll
    for (int kk = 0; kk < 64; kk += 4) {
      v2f a = *(const v2f*)&Pl[(16 * wave + l16) * 68 + kk + 2 * h2];
#pragma unroll
      for (int j = 0; j < 4; ++j) {
        v2f bf = *(const v2f*)&Vt[(16 * j + l16) * 68 + kk + 2 * h2];
        O[j] = WMMA_F32(a, bf, O[j]);
      }
    }
    __syncthreads();   // before next tile overwrites Kl/Vt
  }

  const float epsN = EPSV / (float)NTOK;
#pragma unroll
  for (int j = 0; j < 4; ++j) {
    float cs = colsum[((size_t)b * NUM_HEADS + h) * HD + 16 * j + l16];
#pragma unroll
    for (int r = 0; r < 8; ++r) {
      int q = qt * 64 + 16 * wave + r + 8 * h2;
      float val = (O[j][r] + epsN * cs) / (lrow[r] + EPSV);
      out[((size_t)b * NTOK + q) * CDIM + h * HD + 16 * j + l16] = val;
    }
  }
}

extern "C" void kernel_launch(void* const* d_in, const int* in_sizes, int n_in,
                              void* d_out, int out_size, void* d_ws, size_t ws_size,
                              hipStream_t stream) {
  (void)in_sizes; (void)n_in; (void)out_size; (void)ws_size;
  const float* x      = (const float*)d_in[0];
  const float* dec    = (const float*)d_in[1];
  const float* qkv_w  = (const float*)d_in[2];
  const float* qkv_b  = (const float*)d_in[3];
  const float* proj_w = (const float*)d_in[4];
  const float* proj_b = (const float*)d_in[5];
  float* out = (float*)d_out;

  const int M = 4 * NTOK;                         // 4096
  float* qkvbuf  = (float*)d_ws;                  // 4096 x 2304
  float* attnout = qkvbuf + (size_t)M * 3 * CDIM; // 4096 x 768
  float* csum    = attnout + (size_t)M * CDIM;    // 48 x 64

  // 1) qkv = x @ qkv_w^T + qkv_b
  dim3 g1(3 * CDIM / 64, M / 128);
  gemm_xwt_bias<<<g1, 256, 0, stream>>>(x, qkv_w, qkv_b, qkvbuf, M, 3 * CDIM, CDIM);

  // 2) per-(b,h) V column sums for the exact eps/N term
  v_colsum<<<4 * NUM_HEADS, 64, 0, stream>>>(qkvbuf, csum);

  // 3) policy-masked flash attention
  dim3 g2(NTOK / 64, NUM_HEADS, 4);
  attn_policy<<<g2, 128, 0, stream>>>(qkvbuf, dec, csum, attnout);

  // 4) out = attnout @ proj_w^T + proj_b
  dim3 g3(CDIM / 64, M / 128);
  gemm_xwt_bias<<<g3, 256, 0, stream>>>(attnout, proj_w, proj_b, out, M, CDIM, CDIM);
}